// DistillingLoss_23424751632634
// MI455X (gfx1250) — compile-verified
//
#include <hip/hip_runtime.h>

// Problem size (from reference)
#define BATCH   8192
#define CLASSES 32000
#define N_TOTAL (BATCH * CLASSES)   // 262,144,000 floats per tensor
#define N_VEC4  (N_TOTAL / 4)       // 65,536,000 float4s (divisible)

constexpr int BLOCKS  = 2048;       // 2048 partials -> 8 KB workspace
constexpr int THREADS = 256;        // 8 wave32 per block

typedef __attribute__((ext_vector_type(4))) float fvec4;  // native vector for nontemporal b128 loads
typedef __attribute__((ext_vector_type(2))) float v2f;    // WMMA f32 16x16x4 A/B operand (2 VGPRs)
typedef __attribute__((ext_vector_type(8))) float v8f;    // WMMA f32 16x16x4 C/D accumulator (8 VGPRs)

// ---------------------------------------------------------------------------
// Kernel 1: bandwidth-bound streaming pass.
// Reads 2.1 GB once (non-temporal: streamed, no reuse, don't thrash L2),
// accumulates t * log2(x) per lane (v_log_f32 is hardware log2; the ln2 and
// -1/BATCH factors are applied once at the very end). Deterministic tree
// reduction: wave32 shfl_xor -> LDS -> one partial per block.
// ---------------------------------------------------------------------------
__global__ void __launch_bounds__(THREADS)
distill_loss_partial(const float* __restrict__ x,
                     const float* __restrict__ t,
                     float* __restrict__ partials) {
  const fvec4* __restrict__ x4 = (const fvec4*)x;
  const fvec4* __restrict__ t4 = (const fvec4*)t;

  const int tid    = blockIdx.x * THREADS + threadIdx.x;
  const int stride = gridDim.x * THREADS;   // 524,288 -> exactly 125 iters/thread

  float acc = 0.0f;
#pragma unroll 4
  for (int i = tid; i < N_VEC4; i += stride) {
    fvec4 xv = __builtin_nontemporal_load(&x4[i]);   // global_load_b128, TH=NT
    fvec4 tv = __builtin_nontemporal_load(&t4[i]);
    acc = fmaf(tv.x, __builtin_amdgcn_logf(xv.x), acc);  // v_log_f32 (log2)
    acc = fmaf(tv.y, __builtin_amdgcn_logf(xv.y), acc);
    acc = fmaf(tv.z, __builtin_amdgcn_logf(xv.z), acc);
    acc = fmaf(tv.w, __builtin_amdgcn_logf(xv.w), acc);
  }

  // wave32 butterfly reduction (fixed order -> deterministic)
#pragma unroll
  for (int off = 16; off > 0; off >>= 1)
    acc += __shfl_xor(acc, off, 32);

  __shared__ float sdata[THREADS / 32];
  const int lane = threadIdx.x & 31;
  const int wid  = threadIdx.x >> 5;
  if (lane == 0) sdata[wid] = acc;
  __syncthreads();

  if (threadIdx.x == 0) {
    float s = 0.0f;
#pragma unroll
    for (int w = 0; w < THREADS / 32; ++w) s += sdata[w];
    partials[blockIdx.x] = s;
  }
}

// ---------------------------------------------------------------------------
// Kernel 2: final reduction of 2048 block partials in ONE full wave32
// (EXEC all-1s as WMMA requires). Uses V_WMMA_F32_16X16X4_F32 as a
// K-reduction engine: with B = all-ones, D[m][n] = sum_k A[m][k] + C[m][n],
// so 32 chained WMMAs fold all 2048 values (64 per WMMA: 2 VGPRs x 32 lanes)
// into the f32 accumulator. Then sum the 8 C registers per lane (rows 0-7 in
// lanes 0-15, rows 8-15 in lanes 16-31; all columns identical), one
// shfl_xor(16) to merge the halves, scale by -ln2/BATCH, store the scalar.
// ---------------------------------------------------------------------------
__global__ void __launch_bounds__(32)
distill_loss_final(const float* __restrict__ partials, float* __restrict__ out) {
  const int lane = threadIdx.x;           // 0..31, one full wave
  const v2f* __restrict__ p2 = (const v2f*)partials;

  v2f ones; ones.x = 1.0f; ones.y = 1.0f;
  v8f c = {};                             // zero accumulator

#pragma unroll
  for (int i = 0; i < BLOCKS / 64; ++i) { // 32 WMMAs x 64 values = 2048
    v2f a = p2[i * 32 + lane];
    // D = A x ones + C  -> per-row K-sums accumulate into C
    c = __builtin_amdgcn_wmma_f32_16x16x4_f32(
        /*neg_a=*/false, a, /*neg_b=*/false, ones,
        /*c_mod=*/(short)0, c, /*reuse_a=*/false, /*reuse_b=*/false);
  }

  float s = c[0] + c[1] + c[2] + c[3] + c[4] + c[5] + c[6] + c[7];
  s += __shfl_xor(s, 16, 32);             // merge rows 0-7 with rows 8-15

  if (lane == 0) {
    // loss = -(ln2 / BATCH) * sum(t * log2(x))
    out[0] = s * (-0.69314718055994530942f / (float)BATCH);
  }
}

extern "C" void kernel_launch(void* const* d_in, const int* in_sizes, int n_in,
                              void* d_out, int out_size, void* d_ws, size_t ws_size,
                              hipStream_t stream) {
  const float* x = (const float*)d_in[0];   // input  (strictly positive)
  const float* t = (const float*)d_in[1];   // target
  float* out      = (float*)d_out;          // single f32 scalar
  float* partials = (float*)d_ws;           // 2048 * 4 B = 8 KB scratch

  distill_loss_partial<<<BLOCKS, THREADS, 0, stream>>>(x, t, partials);
  distill_loss_final<<<1, 32, 0, stream>>>(partials, out);
}